// FAGCNConv_68143951118622
// MI455X (gfx1250) — compile-verified
//
#include <hip/hip_runtime.h>
#include <hip/hip_bf16.h>
#include <math.h>

typedef float v2f __attribute__((ext_vector_type(2)));
typedef float v8f __attribute__((ext_vector_type(8)));

#define DIN 256
#define DOUT 256
#define EPS_MIX 0.1f

// ---------------------------------------------------------------------------
// Kernel 1: h = x @ W^T + b   via V_WMMA_F32_16X16X4_F32
// One wave computes a 16x64 output strip (4 adjacent 16x16 C tiles), so the
// A (x) operand is loaded once per k-step and reused across 4 WMMAs:
//   per k-step: 1 A load (v2f) + 4 B loads (v2f) -> 4 WMMAs (2.5 dw/WMMA).
// Layouts (ISA 7.12.2, 32-bit):
//   A (16x4): lane r=lane&15 holds rows; koff=(lane>>4)*2 -> {K+koff, K+koff+1}
//   B (4x16): B[k][n] = W[n][k]  (same striping, N across lanes)
//   C (16x16): c[i] -> row = mbase + (lane>>4)*8 + i, col = nbase + (lane&15)
// ---------------------------------------------------------------------------
__global__ void fagcn_gemm_wmma_f32(const float* __restrict__ x,
                                    const float* __restrict__ W,
                                    const float* __restrict__ bias,
                                    float* __restrict__ h,
                                    int N) {
    const int wavesPerBlock = blockDim.x >> 5;
    const int wave = blockIdx.x * wavesPerBlock + (threadIdx.x >> 5);
    const int ngroups = DOUT / 64;                 // 4 groups of 64 cols
    const int mt = wave / ngroups;                 // 16-row tile index
    const int ng = wave % ngroups;                 // 64-col group index
    if (mt * 16 >= N) return;                      // wave-uniform guard (EXEC stays full)

    const int lane = threadIdx.x & 31;
    const int r    = lane & 15;
    const int koff = (lane >> 4) << 1;

    const float* __restrict__ xrow = x + (size_t)(mt * 16 + r) * DIN;
    const float* __restrict__ wrow0 = W + (size_t)(ng * 64 +  0 + r) * DIN;
    const float* __restrict__ wrow1 = W + (size_t)(ng * 64 + 16 + r) * DIN;
    const float* __restrict__ wrow2 = W + (size_t)(ng * 64 + 32 + r) * DIN;
    const float* __restrict__ wrow3 = W + (size_t)(ng * 64 + 48 + r) * DIN;

    v8f c0 = {}, c1 = {}, c2 = {}, c3 = {};
    #pragma unroll 4
    for (int k = 0; k < DIN; k += 4) {
        v2f a, b0, b1, b2, b3;
        a.x  = xrow[k + koff];   a.y  = xrow[k + koff + 1];
        b0.x = wrow0[k + koff];  b0.y = wrow0[k + koff + 1];
        b1.x = wrow1[k + koff];  b1.y = wrow1[k + koff + 1];
        b2.x = wrow2[k + koff];  b2.y = wrow2[k + koff + 1];
        b3.x = wrow3[k + koff];  b3.y = wrow3[k + koff + 1];
        c0 = __builtin_amdgcn_wmma_f32_16x16x4_f32(false, a, false, b0, (short)0, c0, false, false);
        c1 = __builtin_amdgcn_wmma_f32_16x16x4_f32(false, a, false, b1, (short)0, c1, false, false);
        c2 = __builtin_amdgcn_wmma_f32_16x16x4_f32(false, a, false, b2, (short)0, c2, false, false);
        c3 = __builtin_amdgcn_wmma_f32_16x16x4_f32(false, a, false, b3, (short)0, c3, false, false);
    }

    const int halfsel = lane >> 4;
    const int colbase = ng * 64 + (lane & 15);
    #pragma unroll
    for (int t = 0; t < 4; ++t) {
        const v8f* cp = (t == 0) ? &c0 : (t == 1) ? &c1 : (t == 2) ? &c2 : &c3;
        const int col = colbase + t * 16;
        const float bv = bias[col];
        #pragma unroll
        for (int i = 0; i < 8; ++i) {
            const int row = mt * 16 + halfsel * 8 + i;
            h[(size_t)row * DOUT + col] = (*cp)[i] + bv;
        }
    }
}

// ---------------------------------------------------------------------------
// Kernel 2: per-edge signed attention + scatter-add. One wave32 per edge.
// alpha = tanh(<h[row], h[col]>);  acc[row] += alpha * h[col];  deg[row] += 1
// h fits in L2 (10 MB << 192 MB), so gathers are L2-resident; scatter uses
// native global_atomic_add_f32 (unsafeAtomicAdd).
// ---------------------------------------------------------------------------
__global__ void fagcn_edge_phase(const float* __restrict__ h,
                                 const int* __restrict__ edge_index,
                                 float* __restrict__ acc,
                                 float* __restrict__ deg,
                                 int E) {
    const int wavesPerBlock = blockDim.x >> 5;
    const int e = blockIdx.x * wavesPerBlock + (threadIdx.x >> 5);
    if (e >= E) return;                            // wave-uniform

    const int lane = threadIdx.x & 31;
    const int row = edge_index[e];
    const int col = edge_index[E + e];

    const float4* __restrict__ hr = (const float4*)(h + (size_t)row * DOUT);
    const float4* __restrict__ hc = (const float4*)(h + (size_t)col * DOUT);

    const float4 r0 = hr[lane * 2];
    const float4 r1 = hr[lane * 2 + 1];
    const float4 c0 = hc[lane * 2];
    const float4 c1 = hc[lane * 2 + 1];

    float d = r0.x * c0.x + r0.y * c0.y + r0.z * c0.z + r0.w * c0.w
            + r1.x * c1.x + r1.y * c1.y + r1.z * c1.z + r1.w * c1.w;

    // wave32 all-reduce
    #pragma unroll
    for (int off = 16; off > 0; off >>= 1)
        d += __shfl_xor(d, off, 32);

    const float alpha = tanhf(d);

    float* __restrict__ dst = acc + (size_t)row * DOUT + lane * 8;
    unsafeAtomicAdd(dst + 0, alpha * c0.x);
    unsafeAtomicAdd(dst + 1, alpha * c0.y);
    unsafeAtomicAdd(dst + 2, alpha * c0.z);
    unsafeAtomicAdd(dst + 3, alpha * c0.w);
    unsafeAtomicAdd(dst + 4, alpha * c1.x);
    unsafeAtomicAdd(dst + 5, alpha * c1.y);
    unsafeAtomicAdd(dst + 6, alpha * c1.z);
    unsafeAtomicAdd(dst + 7, alpha * c1.w);

    if (lane == 0) unsafeAtomicAdd(&deg[row], 1.0f);
}

// ---------------------------------------------------------------------------
// Kernel 3: out = EPS*h + (1-EPS) * acc / max(deg, 1)
// ---------------------------------------------------------------------------
__global__ void fagcn_finalize(const float* __restrict__ h,
                               const float* __restrict__ acc,
                               const float* __restrict__ deg,
                               float* __restrict__ out,
                               int total) {
    const int i = blockIdx.x * blockDim.x + threadIdx.x;
    if (i >= total) return;
    const int n = i >> 8;                          // DOUT == 256
    float dg = deg[n];
    dg = dg < 1.0f ? 1.0f : dg;
    out[i] = EPS_MIX * h[i] + (1.0f - EPS_MIX) * (acc[i] / dg);
}

extern "C" void kernel_launch(void* const* d_in, const int* in_sizes, int n_in,
                              void* d_out, int out_size, void* d_ws, size_t ws_size,
                              hipStream_t stream) {
    const float* x  = (const float*)d_in[0];       // [N, 256]
    const int*   ei = (const int*)d_in[1];         // [2, E]
    // d_in[2] edge_weight: unused by the reference
    // d_in[3] num_nodes scalar: N derived from sizes instead
    const float* W  = (const float*)d_in[4];       // [256, 256]
    const float* b  = (const float*)d_in[5];       // [256]

    const int N = in_sizes[0] / DIN;               // 10000
    const int E = in_sizes[1] / 2;                 // 320000

    float* h   = (float*)d_ws;                     // N*256 floats
    float* acc = h + (size_t)N * DOUT;             // N*256 floats
    float* deg = acc + (size_t)N * DOUT;           // N floats

    // zero acc + deg (contiguous); capture-safe async memset
    hipMemsetAsync(acc, 0, ((size_t)N * DOUT + (size_t)N) * sizeof(float), stream);

    // GEMM: (N/16) M-tiles x 4 column-groups, one wave each, 8 waves / block
    {
        const int tiles  = (N / 16) * (DOUT / 64);
        const int blocks = (tiles + 7) / 8;
        fagcn_gemm_wmma_f32<<<blocks, 256, 0, stream>>>(x, W, b, h, N);
    }

    // Edge phase: one wave per edge, 8 waves per block
    {
        const int blocks = (E + 7) / 8;
        fagcn_edge_phase<<<blocks, 256, 0, stream>>>(h, ei, acc, deg, E);
    }

    // Finalize
    {
        const int total  = N * DOUT;
        const int blocks = (total + 255) / 256;
        fagcn_finalize<<<blocks, 256, 0, stream>>>(h, acc, deg, (float*)d_out, total);
    }
}